// Head_84971632984122
// MI455X (gfx1250) — compile-verified
//
#include <hip/hip_runtime.h>
#include <math.h>

// ---------------------------------------------------------------------------
// MI455X (gfx1250) implementation.
//
// Roofline: dominant cost is the gather of 50/500 feature maps per batch row
// (~480 MB HBM traffic => ~21 us @ 23.3 TB/s).  Compute is ~240 MFLOP, i.e.
// ~0.5 FLOP/byte -> memory bound.  We therefore stream with 64-bit loads and
// cast the channel reduction as a per-pixel GEMM via the norm expansion
//   d^2 = ||f||^2 + ||x||^2 - 2 f.x
// so the cross term runs on V_WMMA_F32_16X16X4_F32 (exact f32 FMA, no
// precision loss), leaving the VALU free for the norm accumulation.
//
// wave32 throughout.  No atomics; per-wave running mins combined via LDS.
// ---------------------------------------------------------------------------

typedef __attribute__((ext_vector_type(2))) float v2f;
typedef __attribute__((ext_vector_type(8))) float v8f;

#define WMMA_F32X4(Aop, Bop, Cop)                                              \
  __builtin_amdgcn_wmma_f32_16x16x4_f32(false, (Aop), false, (Bop), (short)0,  \
                                        (Cop), false, false)

__device__ __forceinline__ int iclamp(int v, int lo, int hi) {
  return v < lo ? lo : (v > hi ? hi : v);
}

// ---------------------------------------------------------------------------
// Kernel 1: distance matrix [4,500] via WMMA f32 16x16x4, then per-row
// iterative top-50 selection + mean score.  One workgroup (8 waves).
//
// A (16x4, MxK): lanes 0-15 hold M=lane; VGPR0 = K0 (lanes<16) / K2 (lanes>=16),
// VGPR1 = K1 / K3 (ISA 7.12.2).  B (4x16, KxN) assumed symmetric.
// D (16x16): VGPR j, lanes<16 -> M=j, N=lane.
// ---------------------------------------------------------------------------
__global__ __launch_bounds__(256) void knn_topk_kernel(
    const float* __restrict__ q,     // x2 viewed as [4, 2048]
    const float* __restrict__ bank,  // f2 viewed as [500, 2048]
    float* __restrict__ score,       // d_out[0..3]
    int* __restrict__ idx_out) {     // ws: int[200] = idx[4][50]
  __shared__ float dist[4][512];

  const int wave = threadIdx.x >> 5;
  const int lane = threadIdx.x & 31;
  const int hl   = lane & 15;
  const int chi  = (lane >> 4) << 1;  // 0 for lanes 0-15, 2 for lanes 16-31

  const int mq = (hl < 4) ? hl : 0;   // A rows 4..15 are padding (clamped)
  const float* qa = q + (size_t)mq * 2048 + chi;

  for (int tile = wave; tile < 32; tile += 8) {
    const int n  = tile * 16 + hl;
    const int nc = (n < 500) ? n : 499;
    const float* ba = bank + (size_t)nc * 2048 + chi;

    v8f acc = {0.f, 0.f, 0.f, 0.f, 0.f, 0.f, 0.f, 0.f};
    float qn = 0.f, bn = 0.f;
#pragma unroll 4
    for (int c = 0; c < 2048; c += 4) {
      v2f a  = *(const v2f*)(qa + c);
      v2f bb = *(const v2f*)(ba + c);
      qn = fmaf(a.x, a.x, fmaf(a.y, a.y, qn));
      bn = fmaf(bb.x, bb.x, fmaf(bb.y, bb.y, bn));
      acc = WMMA_F32X4(a, bb, acc);
    }
    // each row's 4 K-values are split across lane L and L+16 -> combine norms
    qn += __shfl_xor(qn, 16, 32);
    bn += __shfl_xor(bn, 16, 32);

    if (lane < 16) {
      const int col = tile * 16 + lane;
#pragma unroll
      for (int j = 0; j < 4; ++j) {            // real batch rows b = 0..3
        const float qnorm = __shfl(qn, j, 32); // q-norm lives at lane j
        const float cross = acc[j];            // M=j, N=lane
        float d2 = qnorm + bn - 2.f * cross;
        d2 = d2 > 0.f ? d2 : 0.f;
        dist[j][col] = (col < 500) ? sqrtf(d2) : __builtin_inff();
      }
    }
  }
  __syncthreads();

  // waves 0..3: iterative top-50 (min + mark) for batch row b = wave
  if (wave < 4) {
    const int b = wave;
    float ssum = 0.f;
    for (int k = 0; k < 50; ++k) {
      float best = __builtin_inff();
      int bi = 0x7fffffff;
      for (int i = lane; i < 512; i += 32) {
        const float v = dist[b][i];
        if (v < best || (v == best && i < bi)) { best = v; bi = i; }
      }
#pragma unroll
      for (int off = 16; off; off >>= 1) {
        const float ov = __shfl_xor(best, off, 32);
        const int   oi = __shfl_xor(bi, off, 32);
        if (ov < best || (ov == best && oi < bi)) { best = ov; bi = oi; }
      }
      dist[b][bi] = __builtin_inff();  // all lanes agree; same-wave LDS in order
      ssum += best;
      if (lane == 0) idx_out[b * 50 + k] = bi;
    }
    if (lane == 0) score[b] = ssum * (1.f / 50.f);
  }
}

// ---------------------------------------------------------------------------
// Kernel 2/3: per-pixel min-over-K squared distance via WMMA.
// For each pixel p: GEMM [rows=200 (idx of all 4 batches, padded to 208), C]
//                 x [C, 16 (4 real batch cols)] in 13 tiles of 16 rows.
// A wave owns (pixel, tile-subset); 4 waves per pixel, 2 pixels per block.
// Output: min d^2 per (b, p).
// ---------------------------------------------------------------------------
template <int HW, int C>
__global__ __launch_bounds__(256) void mask_min_kernel(
    const float* __restrict__ x,    // [4, HW, C]
    const float* __restrict__ f,    // [500, HW, C]
    const int* __restrict__ idxf,   // [200] = idx[4][50]
    float* __restrict__ mout) {     // [4, HW] min squared distance
  __shared__ float part[2][4][4];

  const int wave = threadIdx.x >> 5;
  const int lane = threadIdx.x & 31;
  const int slot = wave >> 2;  // pixel slot within block (0..1)
  const int wsub = wave & 3;   // tile-subset (0..3)
  const int p    = blockIdx.x * 2 + slot;  // grid sized so p < HW always

  const int hl  = lane & 15;
  const int chi = (lane >> 4) << 1;  // c offset 0 / 2 per ISA A layout
  const int bq  = (hl < 4) ? hl : 0;
  const float* xp = x + ((size_t)bq * HW + (size_t)p) * C + chi;

  const int rowoff = (lane >> 4) << 3;  // D-tile M offset: 0 / 8
  const int bcol   = hl;                // this lane's batch column in D

  float runmin = __builtin_inff();

  for (int t = wsub; t < 13; t += 4) {
    const int r  = t * 16 + hl;
    const int rc = (r < 200) ? r : 0;     // padded rows masked out below
    const int n  = idxf[rc];
    const float* fp = f + ((size_t)n * HW + (size_t)p) * C + chi;
    __builtin_prefetch(fp, 0, 1);  // global_prefetch_b8 on the gathered row

    v8f acc = {0.f, 0.f, 0.f, 0.f, 0.f, 0.f, 0.f, 0.f};
    float fn = 0.f, xn = 0.f;
#pragma unroll 4
    for (int c = 0; c < C; c += 4) {
      v2f a  = *(const v2f*)(fp + c);   // A: gathered feature rows
      v2f bb = *(const v2f*)(xp + c);   // B: query features (b = hl)
      fn = fmaf(a.x, a.x, fmaf(a.y, a.y, fn));
      xn = fmaf(bb.x, bb.x, fmaf(bb.y, bb.y, xn));
      acc = WMMA_F32X4(a, bb, acc);
    }
    fn += __shfl_xor(fn, 16, 32);  // row norm (lane hl holds row t*16+hl)
    xn += __shfl_xor(xn, 16, 32);  // query norm for b = hl

#pragma unroll
    for (int j = 0; j < 8; ++j) {
      const int row     = t * 16 + j + rowoff;
      const float fnorm = __shfl(fn, j + rowoff, 32);
      float d2 = fnorm + xn - 2.f * acc[j];
      d2 = d2 > 0.f ? d2 : 0.f;
      // row belongs to batch (row/50); only count rows gathered for bcol
      const bool mine = (bcol < 4) && (row < 200) && ((row / 50) == bcol);
      if (mine && d2 < runmin) runmin = d2;
    }
  }
  // combine the two lane-halves (rows j and j+8 for the same bcol)
  runmin = fminf(runmin, __shfl_xor(runmin, 16, 32));

  if (lane < 4) part[slot][wsub][lane] = runmin;
  __syncthreads();
  if (wsub == 0 && lane < 4) {
    const float m = fminf(fminf(part[slot][0][lane], part[slot][1][lane]),
                          fminf(part[slot][2][lane], part[slot][3][lane]));
    mout[(size_t)lane * HW + p] = m;  // min d^2 (sqrt applied at resize)
  }
}

// ---------------------------------------------------------------------------
// Kernel 4: bilinear upsample (half-pixel centers, edge clamp) of
// sqrt(min d^2) for both levels, averaged -> pre[4,224,224].
// ---------------------------------------------------------------------------
__device__ __forceinline__ float bilin_sqrt(const float* __restrict__ m, int S,
                                            float sy, float sx) {
  const float fy = floorf(sy), fx = floorf(sx);
  const int y0 = (int)fy, x0 = (int)fx;
  const float wy = sy - fy, wx = sx - fx;
  const int y0c = iclamp(y0, 0, S - 1);
  const int y1c = iclamp(y0 + 1, 0, S - 1);
  const int x0c = iclamp(x0, 0, S - 1);
  const int x1c = iclamp(x0 + 1, 0, S - 1);
  const float v00 = sqrtf(m[y0c * S + x0c]);
  const float v01 = sqrtf(m[y0c * S + x1c]);
  const float v10 = sqrtf(m[y1c * S + x0c]);
  const float v11 = sqrtf(m[y1c * S + x1c]);
  const float a = v00 + wx * (v01 - v00);
  const float b = v10 + wx * (v11 - v10);
  return a + wy * (b - a);
}

__global__ __launch_bounds__(256) void resize_combine_kernel(
    const float* __restrict__ m0,   // [4, 28*28] min d^2
    const float* __restrict__ m1,   // [4, 14*14] min d^2
    float* __restrict__ pre) {      // [4, 224, 224]
  const int i = blockIdx.x * blockDim.x + threadIdx.x;  // exactly 4*224*224
  const int x = i % 224;
  const int y = (i / 224) % 224;
  const int b = i / (224 * 224);
  const float v0 = bilin_sqrt(m0 + b * 784, 28, (y + 0.5f) * 0.125f - 0.5f,
                              (x + 0.5f) * 0.125f - 0.5f);
  const float v1 = bilin_sqrt(m1 + b * 196, 14, (y + 0.5f) * 0.0625f - 0.5f,
                              (x + 0.5f) * 0.0625f - 0.5f);
  pre[i] = 0.5f * (v0 + v1);
}

// ---------------------------------------------------------------------------
// Kernels 5/6: separable 33-tap Gaussian (sigma=4), reflect-101 borders.
// Weights computed in-loop and normalized by their running sum (== k1/sum).
// ---------------------------------------------------------------------------
__device__ __forceinline__ int refl101(int t, int n) {
  if (t < 0) t = -t;
  if (t >= n) t = 2 * n - 2 - t;
  return t;
}

__global__ __launch_bounds__(256) void blur_h_kernel(
    const float* __restrict__ in, float* __restrict__ out) {
  const int i = blockIdx.x * blockDim.x + threadIdx.x;
  const int x = i % 224;
  const int base = i - x;  // row base
  float acc = 0.f, wsum = 0.f;
#pragma unroll
  for (int j = -16; j <= 16; ++j) {
    const float w = expf(-(float)(j * j) * (1.f / 32.f));  // 2*sigma^2 = 32
    acc += w * in[base + refl101(x + j, 224)];
    wsum += w;
  }
  out[i] = acc / wsum;
}

__global__ __launch_bounds__(256) void blur_v_kernel(
    const float* __restrict__ in, float* __restrict__ out) {
  const int i = blockIdx.x * blockDim.x + threadIdx.x;
  const int x = i % 224;
  const int y = (i / 224) % 224;
  const int b = i / (224 * 224);
  const float* plane = in + b * 50176;
  float acc = 0.f, wsum = 0.f;
#pragma unroll
  for (int j = -16; j <= 16; ++j) {
    const float w = expf(-(float)(j * j) * (1.f / 32.f));
    acc += w * plane[refl101(y + j, 224) * 224 + x];
    wsum += w;
  }
  out[i] = acc / wsum;
}

// ---------------------------------------------------------------------------
// Workspace layout (bytes); requires ws_size >= ~1.6 MB.
// ---------------------------------------------------------------------------
#define WS_IDX 0          // int[200]
#define WS_M0 1024        // float[4*784]
#define WS_M1 13568       // float[4*196]
#define WS_PRE 16896      // float[4*224*224]
#define WS_TMP 819712     // float[4*224*224]

extern "C" void kernel_launch(void* const* d_in, const int* in_sizes, int n_in,
                              void* d_out, int out_size, void* d_ws,
                              size_t ws_size, hipStream_t stream) {
  (void)in_sizes; (void)n_in; (void)out_size; (void)ws_size;
  const float* x0 = (const float*)d_in[0];  // [4,28,28,512]
  const float* x1 = (const float*)d_in[1];  // [4,14,14,1024]
  const float* x2 = (const float*)d_in[2];  // [4,1,1,2048]
  const float* f0 = (const float*)d_in[3];  // [500,28,28,512]
  const float* f1 = (const float*)d_in[4];  // [500,14,14,1024]
  const float* f2 = (const float*)d_in[5];  // [500,1,1,2048]

  float* out = (float*)d_out;  // [0..3] score, [4..] mask (4*224*224)
  char* ws = (char*)d_ws;
  int*   idx = (int*)(ws + WS_IDX);
  float* m0  = (float*)(ws + WS_M0);
  float* m1  = (float*)(ws + WS_M1);
  float* pre = (float*)(ws + WS_PRE);
  float* tmp = (float*)(ws + WS_TMP);

  knn_topk_kernel<<<1, 256, 0, stream>>>(x2, f2, out, idx);
  // level 0: HW=784 -> 392 blocks * (2 pixels * 4 waves); level 1: HW=196 -> 98
  mask_min_kernel<784, 512><<<392, 256, 0, stream>>>(x0, f0, idx, m0);
  mask_min_kernel<196, 1024><<<98, 256, 0, stream>>>(x1, f1, idx, m1);
  resize_combine_kernel<<<784, 256, 0, stream>>>(m0, m1, pre);  // 784*256 = 200704
  blur_h_kernel<<<784, 256, 0, stream>>>(pre, tmp);
  blur_v_kernel<<<784, 256, 0, stream>>>(tmp, out + 4);
}